// Part_3D_Disnet_28080496181672
// MI455X (gfx1250) — compile-verified
//
#include <hip/hip_runtime.h>

// Part_3D_Disnet concentration loss on MI455X (gfx1250).
// Memory-bound streaming reduction (168 MB @ 23.3 TB/s ~ 7.2us floor).
// Bulk reduction runs on the matrix pipe: V_WMMA_F32_16X16X4_F32 with
// A = 16x4 FP32 data tile, B = 4x16 basis [1, xm, xm^2, 0...].

typedef float v2f __attribute__((ext_vector_type(2)));
typedef float v8f __attribute__((ext_vector_type(8)));

#define LOSS_EPS 1e-8f

constexpr int Bn  = 16;          // batch
constexpr int Pn  = 11;          // parts (last one dropped)
constexpr int PM1 = Pn - 1;      // 10 used channels
constexpr int Hn  = 512;
constexpr int Wn  = 512;
constexpr int NCH = Bn * PM1;    // 160 channels
constexpr int ROWS_PER_WAVE  = 16;
constexpr int WAVES_PER_BLK  = 8;                      // 256 threads
constexpr int BLKS_PER_CH    = Hn / (ROWS_PER_WAVE * WAVES_PER_BLK); // 4

__global__ void zero_ws_kernel(float* __restrict__ ws, int n) {
    int i = blockIdx.x * blockDim.x + threadIdx.x;
    if (i < n) ws[i] = 0.0f;
}

__global__ __launch_bounds__(256) void moment_kernel(
        const float* __restrict__ in, float* __restrict__ ws) {
    const int ch = blockIdx.y;            // 0..159
    const int b  = ch / PM1;
    const int c  = ch % PM1;
    const float* base = in + (size_t)(b * Pn + c) * (size_t)(Hn * Wn);

    const int wave = threadIdx.x >> 5;
    const int lane = threadIdx.x & 31;
    const int n    = lane & 15;           // output column N / A-row within strip
    const int half = lane >> 4;           // K half selector

    const int h0 = (blockIdx.x * WAVES_PER_BLK + wave) * ROWS_PER_WAVE;

    // A operand: lane holds row h0+n, K = half*2 + {0,1} -> contiguous b64.
    const float* aptr = base + (size_t)(h0 + n) * Wn + half * 2;

    // One-hot basis selectors: column N=0 -> 1, N=1 -> xm, N=2 -> xm^2.
    const float alpha = (n == 0) ? 1.0f : 0.0f;
    const float beta  = (n == 1) ? 1.0f : 0.0f;
    const float gamma = (n == 2) ? 1.0f : 0.0f;

    const float inv = 2.0f / (float)Wn;   // 1/256
    float xm0 = (float)(half * 2) * inv - 1.0f;  // col = w + half*2
    float xm1 = xm0 + inv;                       // col + 1

    v8f d0 = {};
    v8f d1 = {};

    // 128 K-chunks of 4 columns; two accumulators break the WMMA dep chain.
    #pragma unroll 2
    for (int kk = 0; kk < Wn / 4; kk += 2) {
        __builtin_prefetch(aptr + 128, 0, 1);   // global_prefetch_b8, 512B ahead

        v2f a0 = *(const v2f*)(aptr);
        v2f b0;
        b0.x = alpha + xm0 * (beta + xm0 * gamma);
        b0.y = alpha + xm1 * (beta + xm1 * gamma);
        d0 = __builtin_amdgcn_wmma_f32_16x16x4_f32(
                 false, a0, false, b0, (short)0, d0, false, false);
        xm0 += 4.0f * inv;
        xm1 += 4.0f * inv;

        v2f a1 = *(const v2f*)(aptr + 4);
        v2f b1;
        b1.x = alpha + xm0 * (beta + xm0 * gamma);
        b1.y = alpha + xm1 * (beta + xm1 * gamma);
        d1 = __builtin_amdgcn_wmma_f32_16x16x4_f32(
                 false, a1, false, b1, (short)0, d1, false, false);
        xm0 += 4.0f * inv;
        xm1 += 4.0f * inv;

        aptr += 8;
    }

    // D layout: this lane holds column N=n for rows M = j + half*8 (j=0..7).
    float t_p = 0.0f, t_y = 0.0f, t_y2 = 0.0f;
    #pragma unroll
    for (int j = 0; j < 8; ++j) {
        const int   row = h0 + j + half * 8;
        const float ym  = (float)row * inv - 1.0f;
        const float v   = d0[j] + d1[j];
        t_p += v;
        t_y  = fmaf(v, ym, t_y);
        t_y2 = fmaf(v, ym * ym, t_y2);
    }

    // Fold lane L with lane L+16 (same N column).
    t_p  += __shfl_down(t_p, 16);
    t_y  += __shfl_down(t_y, 16);
    t_y2 += __shfl_down(t_y2, 16);

    float* acc = ws + ch * 5;   // [S, Sx, Sx2, Sy, Sy2]
    if (lane == 0) {
        atomicAdd(acc + 0, t_p);    // S   (N=0: plain row sums)
        atomicAdd(acc + 3, t_y);    // Sy
        atomicAdd(acc + 4, t_y2);   // Sy2
    } else if (lane == 1) {
        atomicAdd(acc + 1, t_p);    // Sx  (N=1: xm-weighted row sums)
    } else if (lane == 2) {
        atomicAdd(acc + 2, t_p);    // Sx2 (N=2: xm^2-weighted row sums)
    }
}

__global__ __launch_bounds__(256) void finalize_kernel(
        const float* __restrict__ ws, float* __restrict__ out) {
    __shared__ float red[256];
    const int t = threadIdx.x;
    float contrib = 0.0f;
    if (t < NCH) {
        const float* a = ws + t * 5;
        const float S = a[0], Sx = a[1], Sx2 = a[2], Sy = a[3], Sy2 = a[4];
        const float k  = S + LOSS_EPS;
        const float xc = Sx / k;
        const float yc = Sy / k;
        const float vx = (Sx2 - 2.0f * xc * Sx + xc * xc * S) / k;
        const float vy = (Sy2 - 2.0f * yc * Sy + yc * yc * S) / k;
        contrib = vx + vy;
    }
    red[t] = contrib;
    __syncthreads();
    for (int s = 128; s > 0; s >>= 1) {
        if (t < s) red[t] += red[t + s];
        __syncthreads();
    }
    if (t == 0) out[0] = red[0] / (float)Bn;
}

extern "C" void kernel_launch(void* const* d_in, const int* in_sizes, int n_in,
                              void* d_out, int out_size, void* d_ws, size_t ws_size,
                              hipStream_t stream) {
    (void)in_sizes; (void)n_in; (void)out_size; (void)ws_size;
    const float* in  = (const float*)d_in[0];
    float*       ws  = (float*)d_ws;     // 160 * 5 floats of accumulators
    float*       out = (float*)d_out;

    const int nacc = NCH * 5;
    zero_ws_kernel<<<dim3((nacc + 255) / 256), dim3(256), 0, stream>>>(ws, nacc);

    moment_kernel<<<dim3(BLKS_PER_CH, NCH), dim3(256), 0, stream>>>(in, ws);

    finalize_kernel<<<dim3(1), dim3(256), 0, stream>>>(ws, out);
}